// NN_Tag_Pooling_46634754900239
// MI455X (gfx1250) — compile-verified
//
#include <hip/hip_runtime.h>

typedef __attribute__((ext_vector_type(2))) float v2f;
typedef __attribute__((ext_vector_type(8))) float v8f;
typedef unsigned long long u64;

#define N_PEDS 4096
#define TOPK   4

// compare-exchange on packed (distbits<<32)|idx keys, ascending
#define CE(a, b) { u64 mn_ = (a) < (b) ? (a) : (b); u64 mx_ = (a) < (b) ? (b) : (a); (a) = mn_; (b) = mx_; }

// ---------------------------------------------------------------------------
// Kernel 1: top-4 nearest neighbors, ONE WAVE PER PEDESTRIAN.
// 512 blocks x 8 waves -> 4096 waves; each lane scans 128 candidates
// (stride-32 through LDS, conflict-free), keeps a sorted local top-4 of
// packed u64 keys, then 5 shfl_xor bitonic partial-merge steps produce the
// wave-global top-4 (ascending distance; ties -> lower index, matching
// jax.lax.top_k stability).
// ---------------------------------------------------------------------------
__global__ __launch_bounds__(256) void topk_kernel(const float* __restrict__ obs2,
                                                   int* __restrict__ idx_out) {
    __shared__ float2 sp[N_PEDS];                 // 32KB of 320KB/WGP
    const int tid = threadIdx.x;
    const float2* o2 = (const float2*)obs2;
    for (int j = tid; j < N_PEDS; j += 256) sp[j] = o2[j];
    __syncthreads();

    const int lane = tid & 31;
    const int wave = tid >> 5;
    const int i    = blockIdx.x * 8 + wave;       // this wave's pedestrian
    const float2 pi = sp[i];                      // wave-uniform broadcast

    u64 k0 = ~0ull, k1 = ~0ull, k2 = ~0ull, k3 = ~0ull;   // ascending

#pragma unroll 4
    for (int t = 0; t < N_PEDS / 32; ++t) {
        const int j = lane + (t << 5);
        const float2 pj = sp[j];                  // ds_load_b64, conflict-free
        const float dx = pj.x - pi.x;
        const float dy = pj.y - pi.y;
        const float d  = fmaf(dx, dx, fmaf(dy, dy, 1.0f));
        u64 key = ((u64)__float_as_uint(d) << 32) | (unsigned)j;
        key = (j == i) ? ~0ull : key;             // exclude self
        if (key < k3) {                           // wave vote: s_cbranch_execz
            k3 = key;
            if (k3 < k2) { u64 t_ = k2; k2 = k3; k3 = t_; }
            if (k2 < k1) { u64 t_ = k1; k1 = k2; k2 = t_; }
            if (k1 < k0) { u64 t_ = k0; k0 = k1; k1 = t_; }
        }
    }

    // cross-lane reduction: merge two sorted 4-lists -> 4 smallest, 5 steps
#pragma unroll
    for (int off = 16; off >= 1; off >>= 1) {
        const u64 b0 = __shfl_xor(k0, off, 32);
        const u64 b1 = __shfl_xor(k1, off, 32);
        const u64 b2 = __shfl_xor(k2, off, 32);
        const u64 b3 = __shfl_xor(k3, off, 32);
        // min against reversed partner -> 4 smallest as a bitonic sequence
        u64 c0 = k0 < b3 ? k0 : b3;
        u64 c1 = k1 < b2 ? k1 : b2;
        u64 c2 = k2 < b1 ? k2 : b1;
        u64 c3 = k3 < b0 ? k3 : b0;
        // sort the 4-element bitonic sequence ascending
        CE(c0, c2); CE(c1, c3); CE(c0, c1); CE(c2, c3);
        k0 = c0; k1 = c1; k2 = c2; k3 = c3;
    }

    if (lane == 0) {
        idx_out[i * TOPK + 0] = (int)(unsigned)k0;
        idx_out[i * TOPK + 1] = (int)(unsigned)k1;
        idx_out[i * TOPK + 2] = (int)(unsigned)k2;
        idx_out[i * TOPK + 3] = (int)(unsigned)k3;
    }
}

// ---------------------------------------------------------------------------
// Kernel 2: gather neighbor features + ReLU(feat @ W^T + b) via f32 WMMA.
// One wave per 16-row tile of the [16384 x 6] feature matrix (row = 4*ped +
// slot). Two chained V_WMMA_F32_16X16X4_F32 cover K=6 (zero-padded to 8);
// bias preloaded into the accumulator; ReLU on D.
//
// ISA 7.12.2 layouts (wave32):
//   A 16x4:  lane l -> M = l%16;  vgpr v -> K = v + 2*(l/16)
//   B 4x16:  lane l -> N = l%16;  vgpr v -> K = v + 2*(l/16)
//   C/D:     lane l -> N = l%16;  vgpr v -> M = v + 8*(l/16)
// ---------------------------------------------------------------------------
__global__ __launch_bounds__(256) void embed_kernel(const float* __restrict__ obs1,
                                                    const float* __restrict__ obs2,
                                                    const float* __restrict__ W,
                                                    const float* __restrict__ bias,
                                                    const int*   __restrict__ idx,
                                                    float*       __restrict__ out) {
    const int lane    = threadIdx.x & 31;
    const int wave    = threadIdx.x >> 5;
    const int rowBase = (blockIdx.x * 8 + wave) * 16;
    const int n       = lane & 15;                // M row of A / N column of B,C,D
    const int g       = lane >> 4;                // lane half selector

    // ---- this lane's feature row (row = rowBase + n) -----------------------
    const int r = rowBase + n;
    const int p = r >> 2;                         // pedestrian
    const int j = idx[r];                         // neighbor

    const float2 p2 = ((const float2*)obs2)[p];
    const float2 j2 = ((const float2*)obs2)[j];
    const float2 p1 = ((const float2*)obs1)[p];
    const float2 j1 = ((const float2*)obs1)[j];

    const float rpx = j2.x - p2.x;                        // rel_pos
    const float rpy = j2.y - p2.y;
    const float rvx = (j2.x - j1.x) - (p2.x - p1.x);      // rel_vel
    const float rvy = (j2.y - j1.y) - (p2.y - p1.y);

    // feature row f = [rpx, rpy, 1, rvx, rvy, 1, 0, 0]
    v2f a_lo, a_hi;                               // K = 0..3 / K = 4..7
    a_lo.x = g ? 1.0f : rpx;                      // f[2g]
    a_lo.y = g ? rvx  : rpy;                      // f[2g+1]
    a_hi.x = g ? 0.0f : rvy;                      // f[4+2g]
    a_hi.y = g ? 0.0f : 1.0f;                     // f[5+2g]

    // B[K][N] = W[N][K] (N<8, K<6), zero elsewhere; clamp row to stay in-bounds
    const bool  on = (n < 8);
    const float* Wn = W + (n & 7) * 6;
    v2f b_lo, b_hi;
    b_lo.x = on ? Wn[2 * g + 0] : 0.0f;           // K = 2g
    b_lo.y = on ? Wn[2 * g + 1] : 0.0f;           // K = 2g+1
    b_hi.x = (on && !g) ? Wn[4] : 0.0f;           // K = 4   (K=6 -> 0)
    b_hi.y = (on && !g) ? Wn[5] : 0.0f;           // K = 5   (K=7 -> 0)

    const float bb = on ? bias[n & 7] : 0.0f;
    v8f c = { bb, bb, bb, bb, bb, bb, bb, bb };

    c = __builtin_amdgcn_wmma_f32_16x16x4_f32(false, a_lo, false, b_lo,
                                              (short)0, c, false, false);
    c = __builtin_amdgcn_wmma_f32_16x16x4_f32(false, a_hi, false, b_hi,
                                              (short)0, c, false, false);

    // ReLU + store: lane l, vgpr v -> row = rowBase + v + 8g, col = n
    if (on) {
#pragma unroll
        for (int v = 0; v < 8; ++v) {
            const int row = rowBase + v + 8 * g;
            const float e = c[v];
            out[row * 8 + n] = e > 0.0f ? e : 0.0f;   // out = [16384,8] = [4096,32]
        }
    }
}

extern "C" void kernel_launch(void* const* d_in, const int* in_sizes, int n_in,
                              void* d_out, int out_size, void* d_ws, size_t ws_size,
                              hipStream_t stream) {
    const float* obs1 = (const float*)d_in[0];
    const float* obs2 = (const float*)d_in[1];
    const float* W    = (const float*)d_in[2];
    const float* b    = (const float*)d_in[3];
    float* out  = (float*)d_out;
    int*   idxb = (int*)d_ws;                     // 4096*4 ints = 64KB scratch

    // 4096 waves, one per pedestrian: 512 blocks x 8 waves
    topk_kernel<<<N_PEDS / 8, 256, 0, stream>>>(obs2, idxb);

    // 16384 feature rows / (8 waves * 16 rows) = 128 blocks
    embed_kernel<<<(N_PEDS * TOPK) / (8 * 16), 256, 0, stream>>>(
        obs1, obs2, W, b, idxb, out);
}